// ObjectDetectionLoss_36464272343489
// MI455X (gfx1250) — compile-verified
//
#include <hip/hip_runtime.h>
#include <stdint.h>

#define B_ 64
#define N_ 16800
#define C_ 21
#define CHUNKS_ 8
#define CHUNK_ (N_ / CHUNKS_)        // 2100 anchors per block
#define SCALE_XY 10.0f
#define SCALE_WH 5.0f
#define NEG_RATIO 3
#define FP32_EPS 1.1920928955078125e-07f

typedef unsigned int u32;
typedef unsigned long long u64;
typedef u32 v4u __attribute__((ext_vector_type(4)));
typedef int v4i __attribute__((ext_vector_type(4)));
typedef int v8i __attribute__((ext_vector_type(8)));

#if __has_builtin(__builtin_amdgcn_tensor_load_to_lds) && __has_builtin(__builtin_amdgcn_s_wait_tensorcnt)
#define HAVE_TDM 1
#else
#define HAVE_TDM 0
#endif

// ---------------------------------------------------------------------------
// Kernel 1: per-anchor SmoothL1 + CE, labels_neg store, per-(b,chunk) partials
// ---------------------------------------------------------------------------
__global__ __launch_bounds__(256) void k_peranchor(
    const float* __restrict__ p_bboxs,   // [B,N,4]
    const float* __restrict__ g_bboxs,   // [B,N,4]
    const float* __restrict__ p_labels,  // [B,N,C]
    const int*   __restrict__ g_labels,  // [B,N]
    const float* __restrict__ ancs,      // [N,4]
    float* __restrict__ labels_neg,      // [B,N]
    float* __restrict__ lb_part,         // [B,CHUNKS]
    float* __restrict__ cep_part,        // [B,CHUNKS]
    int*   __restrict__ pn_part)         // [B,CHUNKS]
{
  const int b     = blockIdx.y;
  const int chunk = blockIdx.x;
  const int n0    = chunk * CHUNK_;
  const int tid   = threadIdx.x;

  __shared__ float sAnc[CHUNK_ * 4];     // 33.6 KB anchor stage
  __shared__ float sLB[256];
  __shared__ float sCE[256];
  __shared__ int   sPN[256];

#if HAVE_TDM
  if (tid < 32) {  // wave 0 issues one TDM descriptor for the whole tile
    u64 gaddr  = (u64)(uintptr_t)(ancs + (size_t)n0 * 4);
    u32 ldsoff = (u32)(uintptr_t)(&sAnc[0]);   // low 32 bits of LDS generic addr
    const u32 elems = CHUNK_ * 4;              // 8400 x 4B elements, 1 row
    v4u g0;
    g0.x = 1u;                                 // count=1 (valid), no gather
    g0.y = ldsoff;                             // lds_addr
    g0.z = (u32)gaddr;                         // global_addr[31:0]
    g0.w = (u32)((gaddr >> 32) & 0x01FFFFFFu) | (2u << 30);  // addr[56:32] | type=2
    v8i g1;
    g1[0] = (int)(2u << 16);                   // data_size=4B, no barrier/iter/pad
    g1[1] = (int)(elems << 16);                // tensor_dim0 lo16 (abar addr = 0)
    g1[2] = (int)(1u << 16);                   // tensor_dim0 hi=0 | tensor_dim1=1
    g1[3] = (int)(elems << 16);                // tensor_dim1 hi=0 | tile_dim0
    g1[4] = (int)1u;                           // tile_dim1=1, tile_dim2=0
    g1[5] = (int)elems;                        // tensor_dim0_stride lo32
    g1[6] = (int)(elems << 16);                // stride0 hi=0 | stride1 lo16
    g1[7] = 0;
    v4i z4 = {0, 0, 0, 0};
    v8i z8 = {0, 0, 0, 0, 0, 0, 0, 0};
    __builtin_amdgcn_tensor_load_to_lds(g0, g1, z4, z4, z8, 0);
    __builtin_amdgcn_s_wait_tensorcnt(0);
  }
#else
  for (int i = tid; i < CHUNK_ * 4; i += 256) sAnc[i] = ancs[(size_t)n0 * 4 + i];
#endif
  __syncthreads();

  const float4* pb4 = (const float4*)p_bboxs;   // 16B-aligned per-anchor rows
  const float4* gb4 = (const float4*)g_bboxs;

  float lb  = 0.0f;   // sum sl1 over positives
  float cep = 0.0f;   // sum CE over positives
  int   pn  = 0;      // positive count

  for (int n = n0 + tid; n < n0 + CHUNK_; n += 256) {
    const float* a = &sAnc[(size_t)(n - n0) * 4];
    const float ax = a[0], ay = a[1], aw = a[2], ah = a[3];
    const size_t idx = (size_t)b * N_ + n;

    float4 g = gb4[idx];
    float4 p = pb4[idx];
    float tx = SCALE_XY * (g.x - ax) / aw;
    float ty = SCALE_XY * (g.y - ay) / ah;
    float tw = SCALE_WH * logf(g.z / aw);
    float th = SCALE_WH * logf(g.w / ah);

    float sl1 = 0.0f;
    {
      float d, ad;
      d = p.x - tx; ad = fabsf(d); sl1 += (ad < 1.0f) ? 0.5f * d * d : ad - 0.5f;
      d = p.y - ty; ad = fabsf(d); sl1 += (ad < 1.0f) ? 0.5f * d * d : ad - 0.5f;
      d = p.z - tw; ad = fabsf(d); sl1 += (ad < 1.0f) ? 0.5f * d * d : ad - 0.5f;
      d = p.w - th; ad = fabsf(d); sl1 += (ad < 1.0f) ? 0.5f * d * d : ad - 0.5f;
    }

    const int lab = g_labels[idx];
    const bool pos = lab > 0;

    const float* xr = p_labels + idx * C_;
    __builtin_prefetch((const void*)(xr + (size_t)256 * C_), 0, 0);
    float xs[C_];
#pragma unroll
    for (int c = 0; c < C_; ++c) xs[c] = xr[c];
    float m = xs[0];
#pragma unroll
    for (int c = 1; c < C_; ++c) m = fmaxf(m, xs[c]);
    float s = 0.0f, xl = xs[0];
#pragma unroll
    for (int c = 0; c < C_; ++c) {
      s += expf(xs[c] - m);
      xl = (c == lab) ? xs[c] : xl;
    }
    const float ce = m + logf(s) - xl;   // = -log_softmax[label]

    labels_neg[idx] = pos ? 0.0f : ce;
    if (pos) { lb += sl1; cep += ce; ++pn; }
  }

  sLB[tid] = lb; sCE[tid] = cep; sPN[tid] = pn;
  __syncthreads();
  for (int off = 128; off > 0; off >>= 1) {
    if (tid < off) {
      sLB[tid] += sLB[tid + off];
      sCE[tid] += sCE[tid + off];
      sPN[tid] += sPN[tid + off];
    }
    __syncthreads();
  }
  if (tid == 0) {
    lb_part[b * CHUNKS_ + chunk]  = sLB[0];
    cep_part[b * CHUNKS_ + chunk] = sCE[0];
    pn_part[b * CHUNKS_ + chunk]  = sPN[0];
  }
}

// ---------------------------------------------------------------------------
// Kernel 2: per-batch exact radix-select of k-th largest CE + top-k sum
// ---------------------------------------------------------------------------
__global__ __launch_bounds__(256) void k_select(
    const float* __restrict__ labels_neg,  // [B,N]
    const float* __restrict__ lb_part,
    const float* __restrict__ cep_part,
    const int*   __restrict__ pn_part,
    float* __restrict__ lb_out,            // [B]
    float* __restrict__ ll_out,            // [B]
    int*   __restrict__ pn_out)            // [B]
{
  const int b = blockIdx.x;
  const int tid = threadIdx.x;

  __shared__ int   hist[256];
  __shared__ float rsum[256];
  __shared__ int   rcnt[256];
  __shared__ int   s_k, s_rem;
  __shared__ u32   s_pref;
  __shared__ float s_cep;

  if (tid == 0) {
    float lb = 0.0f, cep = 0.0f; int pn = 0;
    for (int c = 0; c < CHUNKS_; ++c) {
      lb  += lb_part[b * CHUNKS_ + c];
      cep += cep_part[b * CHUNKS_ + c];
      pn  += pn_part[b * CHUNKS_ + c];
    }
    lb_out[b] = lb;
    pn_out[b] = pn;
    int k = NEG_RATIO * pn; if (k > N_) k = N_;
    s_k = k; s_rem = k; s_pref = 0u; s_cep = cep;
  }
  __syncthreads();

  const float* row = labels_neg + (size_t)b * N_;
  const int k = s_k;

  if (k > 0) {
    // 4 passes of 8-bit radix select for the k-th largest (all values >= 0,
    // so uint bit order == float order)
    for (int shift = 24; shift >= 0; shift -= 8) {
      hist[tid] = 0;
      __syncthreads();
      const u32 hmask = (shift == 24) ? 0u : (0xFFFFFFFFu << (shift + 8));
      const u32 pref = s_pref;
      for (int n = tid; n < N_; n += 256) {
        u32 u = __float_as_uint(row[n]);
        if ((u & hmask) == pref) atomicAdd(&hist[(u >> shift) & 255], 1);
      }
      __syncthreads();
      if (tid == 0) {
        int rem = s_rem, bsel = 0;
        for (int v = 255; v >= 0; --v) {
          int c = hist[v];
          if (c < rem) rem -= c; else { bsel = v; break; }
        }
        s_rem = rem;
        s_pref = pref | ((u32)bsel << shift);
      }
      __syncthreads();
    }

    const u32 T = s_pref;                 // exact bit pattern of k-th largest
    const float tval = __uint_as_float(T);
    float sgt = 0.0f; int cgt = 0;
    for (int n = tid; n < N_; n += 256) {
      float v = row[n];
      if (__float_as_uint(v) > T) { sgt += v; ++cgt; }
    }
    rsum[tid] = sgt; rcnt[tid] = cgt;
    __syncthreads();
    for (int off = 128; off > 0; off >>= 1) {
      if (tid < off) { rsum[tid] += rsum[tid + off]; rcnt[tid] += rcnt[tid + off]; }
      __syncthreads();
    }
    if (tid == 0) {
      // ties at T are all equal -> sum is exact under stable-sort semantics
      float topk = rsum[0] + (float)(k - rcnt[0]) * tval;
      ll_out[b] = s_cep + topk;
    }
  } else {
    if (tid == 0) ll_out[b] = s_cep;      // pos_num==0 row is masked out anyway
  }
}

// ---------------------------------------------------------------------------
// Kernel 3: final three means
// ---------------------------------------------------------------------------
__global__ __launch_bounds__(64) void k_final(
    const float* __restrict__ lb_out,
    const float* __restrict__ ll_out,
    const int*   __restrict__ pn_out,
    float* __restrict__ out)               // [3]
{
  __shared__ float st[B_], sb[B_], sl[B_];
  const int t = threadIdx.x;
  if (t < B_) {
    const int p = pn_out[t];
    const float mask = (p > 0) ? 1.0f : 0.0f;
    const float pf = fmaxf((float)p, FP32_EPS);
    const float inv = mask / pf;
    st[t] = (lb_out[t] + ll_out[t]) * inv;   // COEFF = (1,1)
    sb[t] = lb_out[t] * inv;
    sl[t] = ll_out[t] * inv;
  }
  __syncthreads();
  if (t == 0) {
    float s0 = 0.0f, s1 = 0.0f, s2 = 0.0f;
    for (int i = 0; i < B_; ++i) { s0 += st[i]; s1 += sb[i]; s2 += sl[i]; }
    out[0] = s0 / (float)B_;
    out[1] = s1 / (float)B_;
    out[2] = s2 / (float)B_;
  }
}

// ---------------------------------------------------------------------------
extern "C" void kernel_launch(void* const* d_in, const int* in_sizes, int n_in,
                              void* d_out, int out_size, void* d_ws, size_t ws_size,
                              hipStream_t stream) {
  const float* p_bboxs  = (const float*)d_in[0];
  const float* g_bboxs  = (const float*)d_in[1];
  const float* p_labels = (const float*)d_in[2];
  const int*   g_labels = (const int*)d_in[3];
  const float* ancs     = (const float*)d_in[4];

  char* ws = (char*)d_ws;
  size_t off = 0;
  float* labels_neg = (float*)(ws + off); off += (size_t)B_ * N_ * sizeof(float);
  float* lb_part    = (float*)(ws + off); off += (size_t)B_ * CHUNKS_ * sizeof(float);
  float* cep_part   = (float*)(ws + off); off += (size_t)B_ * CHUNKS_ * sizeof(float);
  int*   pn_part    = (int*)  (ws + off); off += (size_t)B_ * CHUNKS_ * sizeof(int);
  float* lb_out     = (float*)(ws + off); off += (size_t)B_ * sizeof(float);
  float* ll_out     = (float*)(ws + off); off += (size_t)B_ * sizeof(float);
  int*   pn_out     = (int*)  (ws + off); off += (size_t)B_ * sizeof(int);
  (void)in_sizes; (void)n_in; (void)out_size; (void)ws_size;

  dim3 grid1(CHUNKS_, B_);
  k_peranchor<<<grid1, 256, 0, stream>>>(p_bboxs, g_bboxs, p_labels, g_labels,
                                         ancs, labels_neg, lb_part, cep_part,
                                         pn_part);
  k_select<<<B_, 256, 0, stream>>>(labels_neg, lb_part, cep_part, pn_part,
                                   lb_out, ll_out, pn_out);
  k_final<<<1, 64, 0, stream>>>(lb_out, ll_out, pn_out, (float*)d_out);
}